// CCLayer_65944927863381
// MI455X (gfx1250) — compile-verified
//
#include <hip/hip_runtime.h>

typedef __attribute__((ext_vector_type(2))) float v2f;
typedef __attribute__((ext_vector_type(8))) float v8f;

#define D_DIM 256
#define P_DIM 32
#define N_DIM 262144

#define TILES_PER_BLOCK 4   // each block: 4 groups of 128 columns = 512 columns

// t-matrix register access helpers (indices are compile-time after unroll):
// t row m (0..31) lives in t0[m&7] (m<16) or t1[m&7] (m>=16),
// at source lane (col) + 16*((m&15)>>3).
#define TREG(m) (((m) < 16) ? t0[(m) & 7] : t1[(m) & 7])
#define TROW(m) (__shfl(TREG(m), l + 16 * (((m) & 15) >> 3), 32))

__global__ __launch_bounds__(256) void cc_layer_wmma(
    const float* __restrict__ ZPi, const float* __restrict__ U,
    const float* __restrict__ alpha, float* __restrict__ out)
{
    // A-fragment packs of U, in exact WMMA f32 16x16x4 operand layout.
    __shared__ float pack1[64 * 2 * 32 * 2];  // 32 KB : GEMM1 (A = U^T), [k][ptile][lane]{c0,c1}
    __shared__ float pack2[16 * 8 * 32 * 2];  // 32 KB : GEMM2 (A = U),   [dtile][k2][lane]{c0,c1}

    const int tid  = threadIdx.x;
    const int lane = tid & 31;
    const int wave = tid >> 5;
    const int l    = lane & 15;   // column / M index within half-wave
    const int hi   = lane >> 4;   // half-wave select (K offset +2 / M offset +8)

    // ---------------- Phase 0: pack U into LDS fragment layouts (once per block) ------------
    for (int i = tid; i < 8192; i += 256) {
        int c  = i & 1;
        int ln = (i >> 1) & 31;
        int pt = (i >> 6) & 1;
        int k  = i >> 7;
        int row = 4 * k + 2 * (ln >> 4) + c;      // K index into U^T  -> row of U
        int col = pt * 16 + (ln & 15);            // M index of U^T    -> col of U
        pack1[i] = U[row * P_DIM + col];
    }
    for (int i = tid; i < 8192; i += 256) {
        int c  = i & 1;
        int ln = (i >> 1) & 31;
        int k2 = (i >> 6) & 7;
        int dt = i >> 9;
        int row = dt * 16 + (ln & 15);            // M index of U (row)
        int col = 4 * k2 + 2 * (ln >> 4) + c;     // K index of U (col)
        pack2[i] = U[row * P_DIM + col];
    }
    __syncthreads();

    // alpha values for this lane's (v, hi) positions are tile-invariant: hoist to registers.
    float a_lo[8], a_hi[8];
    #pragma unroll
    for (int v = 0; v < 8; ++v) {
        a_lo[v] = alpha[v + 8 * hi];
        a_hi[v] = alpha[16 + v + 8 * hi];
    }

    for (int t = 0; t < TILES_PER_BLOCK; ++t) {
        const long n0  = (long)blockIdx.x * (128 * TILES_PER_BLOCK) + (long)t * 128
                       + (long)wave * 16;
        const long col = n0 + l;

        // ------------- GEMM1: UtZ = U^T (32x256) @ Z(:,tile) (256x16) -------------
        v8f acc0 = {};   // rows 0..15 of UtZ
        v8f acc1 = {};   // rows 16..31 of UtZ
        {
            const float* zb = ZPi + col;
            #pragma unroll 8
            for (int k = 0; k < 64; ++k) {
                int r = 4 * k + 2 * hi;
                v2f b = { zb[(long)r * N_DIM], zb[(long)(r + 1) * N_DIM] };
                v2f a0 = *(const v2f*)&pack1[(k * 2 + 0) * 64 + lane * 2];
                v2f a1 = *(const v2f*)&pack1[(k * 2 + 1) * 64 + lane * 2];
                acc0 = __builtin_amdgcn_wmma_f32_16x16x4_f32(false, a0, false, b,
                                                             (short)0, acc0, false, false);
                acc1 = __builtin_amdgcn_wmma_f32_16x16x4_f32(false, a1, false, b,
                                                             (short)0, acc1, false, false);
            }
        }

        // ------------- Pi load (C-layout), sum_pi, t = (UtZ - alpha) * Pi -------------
        float pi0[8], pi1[8], t0[8], t1[8];
        float s = 0.f;
        #pragma unroll
        for (int v = 0; v < 8; ++v) {
            pi0[v] = ZPi[(long)(D_DIM + v + 8 * hi) * N_DIM + col];
            pi1[v] = ZPi[(long)(D_DIM + 16 + v + 8 * hi) * N_DIM + col];
            s += pi0[v] + pi1[v];
        }
        float sum_pi = s + __shfl_xor(s, 16, 32);  // lanes l and l+16 hold complementary rows

        #pragma unroll
        for (int v = 0; v < 8; ++v) {
            t0[v] = (acc0[v] - a_lo[v]) * pi0[v];
            t1[v] = (acc1[v] - a_hi[v]) * pi1[v];
        }

        // ------------- Re-layout t (C-layout -> B-operand layout) via lane permutes -------------
        v2f b2[8];
        #pragma unroll
        for (int k2 = 0; k2 < 8; ++k2) {
            const int m0 = 4 * k2;       // K row for element 0 (hi==0); +2 when hi==1
            const int m1 = 4 * k2 + 1;   // K row for element 1
            float xlo = TROW(m0);
            float xhi = TROW(m0 + 2);
            float ylo = TROW(m1);
            float yhi = TROW(m1 + 2);
            b2[k2] = (v2f){ hi ? xhi : xlo, hi ? yhi : ylo };
        }

        // ------------- GEMM2: U (256x32) @ t (32x16), fused combine + NT store -------------
        #pragma unroll 2
        for (int dt = 0; dt < 16; ++dt) {
            v8f acc = {};
            #pragma unroll
            for (int k2 = 0; k2 < 8; ++k2) {
                v2f a = *(const v2f*)&pack2[(dt * 8 + k2) * 64 + lane * 2];
                acc = __builtin_amdgcn_wmma_f32_16x16x4_f32(false, a, false, b2[k2],
                                                            (short)0, acc, false, false);
            }
            #pragma unroll
            for (int v = 0; v < 8; ++v) {
                long row = (long)(dt * 16 + v + 8 * hi);
                float z = ZPi[row * N_DIM + col];   // L2/WGP$ hit (streamed in GEMM1)
                __builtin_nontemporal_store(z * sum_pi - acc[v], &out[row * N_DIM + col]);
            }
        }

        // ------------- Pi passthrough rows 256..287 (streaming store) -------------
        #pragma unroll
        for (int v = 0; v < 8; ++v) {
            __builtin_nontemporal_store(pi0[v],
                &out[(long)(D_DIM + v + 8 * hi) * N_DIM + col]);
            __builtin_nontemporal_store(pi1[v],
                &out[(long)(D_DIM + 16 + v + 8 * hi) * N_DIM + col]);
        }
    }
}

extern "C" void kernel_launch(void* const* d_in, const int* in_sizes, int n_in,
                              void* d_out, int out_size, void* d_ws, size_t ws_size,
                              hipStream_t stream) {
    (void)in_sizes; (void)n_in; (void)out_size; (void)d_ws; (void)ws_size;
    const float* ZPi   = (const float*)d_in[0];
    const float* U     = (const float*)d_in[1];
    const float* alpha = (const float*)d_in[2];
    float* out = (float*)d_out;

    dim3 grid(N_DIM / (128 * TILES_PER_BLOCK));  // 512 blocks x 512 columns
    dim3 block(256);
    cc_layer_wmma<<<grid, block, 0, stream>>>(ZPi, U, alpha, out);
}